// DynamicBalanceLoss_v3_1992864825405
// MI455X (gfx1250) — compile-verified
//
#include <hip/hip_runtime.h>
#include <math.h>

// Dynamic-balance loss for MI455X (gfx1250, wave32).
// Bandwidth-bound: 1.05 GB streamed once from HBM (~45us floor @ 23.3 TB/s).
// Single-pass online softmax + argmax per row; 8x-unrolled NT b128 streaming
// loads (8 outstanding vmem ops per wave ~ 4KB in flight per wave);
// wave32 shuffle reductions; deterministic two-kernel finalize.

typedef float v4f __attribute__((ext_vector_type(4)));

#define LOG2E 1.4426950408889634f
#define LN2   0.6931471805599453f

__device__ __forceinline__ float fast_exp2(float x) { return __builtin_amdgcn_exp2f(x); }
__device__ __forceinline__ float fast_log2(float x) { return __builtin_amdgcn_logf(x); }

__device__ __forceinline__ float max4(v4f v) {
  return fmaxf(fmaxf(v.x, v.y), fmaxf(v.z, v.w));
}
// first element index (0..3) within chunk equal to mv
__device__ __forceinline__ int sel4(v4f v, float mv) {
  return (v.x == mv) ? 0 : (v.y == mv) ? 1 : (v.z == mv) ? 2 : 3;
}
// sum of exp(v[i] - m): one FMA + one v_exp_f32 per element; ml = m*LOG2E
__device__ __forceinline__ float esum4(v4f v, float ml) {
  float e0 = fast_exp2(__builtin_fmaf(v.x, LOG2E, -ml));
  float e1 = fast_exp2(__builtin_fmaf(v.y, LOG2E, -ml));
  float e2 = fast_exp2(__builtin_fmaf(v.z, LOG2E, -ml));
  float e3 = fast_exp2(__builtin_fmaf(v.w, LOG2E, -ml));
  return (e0 + e1) + (e2 + e3);
}

// Merge two online-softmax/argmax triples; prefers lower index on max ties
// (matches jnp.argmax first-occurrence semantics).
__device__ __forceinline__ void merge3(float& m, int& i, float& s,
                                       float mo, int io, float so) {
  bool ow = (mo > m) || ((mo == m) && (io < i));
  float mh = ow ? mo : m;
  float ml = ow ? m  : mo;
  float sh = ow ? so : s;
  float sl = ow ? s  : so;
  int   ih = ow ? io : i;
  s = sh + sl * fast_exp2((ml - mh) * LOG2E);
  m = mh;
  i = ih;
}

__global__ __launch_bounds__(256)
void dbl_row_kernel(const float* __restrict__ logits,
                    const int*   __restrict__ target,
                    const int*   __restrict__ ew_true,
                    const int*   __restrict__ ew_pred,
                    const float* __restrict__ ew_w,
                    float* __restrict__ ws_l1,
                    float* __restrict__ ws_l2,
                    float* __restrict__ ws_c1,
                    int C) {
  const int row = blockIdx.x;
  const int NV  = C >> 2;                       // float4 chunks per row
  const float* rp = logits + (size_t)row * (size_t)C;
  const v4f*   vp = (const v4f*)rp;
  const int tid = threadIdx.x;

  float m  = -INFINITY;   // running max
  float s  = 0.0f;        // running sum of exp(x - m)
  int   am = 0;           // running argmax (column index)

  // ---- main loop: 8 independent NT b128 loads in flight per wave ----
  int jb = 0;
  for (; jb + 2048 <= NV; jb += 2048) {
    const int j0 = jb + tid;
    v4f v[8];
    #pragma unroll
    for (int u = 0; u < 8; ++u)
      v[u] = __builtin_nontemporal_load(vp + j0 + 256 * u);  // imm offsets 0..28KB

    float mu[8];
    #pragma unroll
    for (int u = 0; u < 8; ++u) mu[u] = max4(v[u]);
    const float mg = fmaxf(fmaxf(fmaxf(mu[0], mu[1]), fmaxf(mu[2], mu[3])),
                           fmaxf(fmaxf(mu[4], mu[5]), fmaxf(mu[6], mu[7])));
    if (mg > m) {                                // rare: new record for this lane
      s *= fast_exp2((m - mg) * LOG2E);
      int idx = 0;
      #pragma unroll
      for (int u = 7; u >= 0; --u)               // downward scan: first match wins
        idx = (mu[u] == mg) ? (((j0 + 256 * u) << 2) + sel4(v[u], mg)) : idx;
      am = idx;
      m = mg;
    }
    const float ml = m * LOG2E;
    // 32 independent v_exp_f32, balanced tree -> single add into s per iter
    s += ((esum4(v[0], ml) + esum4(v[1], ml)) + (esum4(v[2], ml) + esum4(v[3], ml)))
       + ((esum4(v[4], ml) + esum4(v[5], ml)) + (esum4(v[6], ml) + esum4(v[7], ml)));
  }
  // ---- mid loop: 4 loads in flight ----
  for (; jb + 1024 <= NV; jb += 1024) {
    const int j0 = jb + tid;
    v4f a = __builtin_nontemporal_load(vp + j0);
    v4f b = __builtin_nontemporal_load(vp + j0 + 256);
    v4f c = __builtin_nontemporal_load(vp + j0 + 512);
    v4f d = __builtin_nontemporal_load(vp + j0 + 768);
    float mA = max4(a), mB = max4(b), mC = max4(c), mD = max4(d);
    float mg = fmaxf(fmaxf(mA, mB), fmaxf(mC, mD));
    if (mg > m) {
      s *= fast_exp2((m - mg) * LOG2E);
      const int base = j0 << 2;
      am = (mA == mg) ? (base        + sel4(a, mg))
         : (mB == mg) ? (base + 1024 + sel4(b, mg))
         : (mC == mg) ? (base + 2048 + sel4(c, mg))
         :               (base + 3072 + sel4(d, mg));
      m = mg;
    }
    const float ml = m * LOG2E;
    s += (esum4(a, ml) + esum4(b, ml)) + (esum4(c, ml) + esum4(d, ml));
  }
  // ---- remainder chunks ----
  for (int j = jb + tid; j < NV; j += 256) {
    v4f v = __builtin_nontemporal_load(vp + j);
    float m4 = max4(v);
    if (m4 > m) {
      s *= fast_exp2((m - m4) * LOG2E);
      am = (j << 2) + sel4(v, m4);
      m = m4;
    }
    s += esum4(v, m * LOG2E);
  }

  // wave32 intra-wave reduction (lane 0 valid)
  #pragma unroll
  for (int off = 16; off > 0; off >>= 1) {
    float mo = __shfl_down(m,  off, 32);
    float so = __shfl_down(s,  off, 32);
    int   io = __shfl_down(am, off, 32);
    merge3(m, am, s, mo, io, so);
  }

  // cross-wave (8 waves per 256-thread block) via LDS
  __shared__ float sm[8], ss[8];
  __shared__ int   si[8];
  const int wave = tid >> 5;
  if ((tid & 31) == 0) { sm[wave] = m; ss[wave] = s; si[wave] = am; }
  __syncthreads();

  if (tid == 0) {
    #pragma unroll
    for (int w = 1; w < 8; ++w) merge3(m, am, s, sm[w], si[w], ss[w]);

    const int   t    = target[row];
    const float xt   = rp[t];
    const float loss = (m - xt) + fast_log2(s) * LN2;   // -log_softmax[target]

    bool  in_dict = false;
    float wgt     = 0.0f;
    #pragma unroll
    for (int e = 0; e < 3; ++e) {
      const bool match = (t == ew_true[e]) && (am == ew_pred[e]);
      in_dict = in_dict || match;
      wgt += match ? ew_w[e] : 0.0f;
    }
    ws_l1[row] = in_dict ? loss * wgt : 0.0f;   // contributes to loss1_sum
    ws_l2[row] = in_dict ? 0.0f : loss;         // contributes to loss2_sum
    ws_c1[row] = in_dict ? 1.0f : 0.0f;         // contributes to count1
  }
}

__global__ __launch_bounds__(1024)
void dbl_finalize_kernel(const float* __restrict__ ws_l1,
                         const float* __restrict__ ws_l2,
                         const float* __restrict__ ws_c1,
                         float* __restrict__ out, int B) {
  const int tid = threadIdx.x;
  float s1 = 0.0f, s2 = 0.0f, s3 = 0.0f;
  for (int i = tid; i < B; i += 1024) {
    s1 += ws_l1[i];
    s2 += ws_l2[i];
    s3 += ws_c1[i];
  }
  #pragma unroll
  for (int off = 16; off > 0; off >>= 1) {
    s1 += __shfl_down(s1, off, 32);
    s2 += __shfl_down(s2, off, 32);
    s3 += __shfl_down(s3, off, 32);
  }
  __shared__ float a1[32], a2[32], a3[32];
  const int wave = tid >> 5;
  if ((tid & 31) == 0) { a1[wave] = s1; a2[wave] = s2; a3[wave] = s3; }
  __syncthreads();

  if (tid == 0) {
    for (int w = 1; w < 32; ++w) { s1 += a1[w]; s2 += a2[w]; s3 += a3[w]; }
    const float Bf = (float)B;
    // faithful to reference: divide only if the corresponding count > 0
    const float loss1 = (s3 > 0.0f)        ? (s1 / Bf) : s1;
    const float loss2 = ((Bf - s3) > 0.0f) ? (s2 / Bf) : s2;
    const float K = 0.5f, T = 3.0f;
    const float wl2 = 1.0f / (1.0f + fast_exp2(-(K * (loss1 - T)) * LOG2E));
    out[0] = (loss1 > 0.0f) ? (loss1 + wl2 * loss2) : (loss2 / Bf);
  }
}

extern "C" void kernel_launch(void* const* d_in, const int* in_sizes, int n_in,
                              void* d_out, int out_size, void* d_ws, size_t ws_size,
                              hipStream_t stream) {
  const float* logits  = (const float*)d_in[0];   // [B, C] fp32
  const int*   target  = (const int*)d_in[1];     // [B] int
  const int*   ew_true = (const int*)d_in[2];     // [3]
  const int*   ew_pred = (const int*)d_in[3];     // [3]
  const float* ew_w    = (const float*)d_in[4];   // [3]

  const int B = in_sizes[1];
  const int C = in_sizes[0] / B;

  // d_ws: three [B] float arrays, fully rewritten every call (no init needed).
  float* ws_l1 = (float*)d_ws;
  float* ws_l2 = ws_l1 + B;
  float* ws_c1 = ws_l2 + B;

  dbl_row_kernel<<<B, 256, 0, stream>>>(logits, target, ew_true, ew_pred, ew_w,
                                        ws_l1, ws_l2, ws_c1, C);
  dbl_finalize_kernel<<<1, 1024, 0, stream>>>(ws_l1, ws_l2, ws_c1,
                                              (float*)d_out, B);
}